// ConexaoRegional_32564442038402
// MI455X (gfx1250) — compile-verified
//
#include <hip/hip_runtime.h>

typedef __attribute__((ext_vector_type(2))) float v2f;
typedef __attribute__((ext_vector_type(8))) float v8f;

#define B_   64
#define H_   128
#define W_   128
#define R_   4
#define GW_  32
#define N_   1024
#define DS_  512            // D*S = 16*32
#define OUTROW_ ((size_t)N_ * DS_)   // stride between consecutive b in out

// out[b, n, ds] = sum_r x[b, 0, i*4+r, j*4+r] * pesos[n*8192 + ds*16 + r*5]
// Per region n: [64x4] @ [4x512] GEMM, K=4 -> V_WMMA_F32_16X16X4_F32 tiles.
__global__ __launch_bounds__(256) void conexao_wmma_kernel(
    const float* __restrict__ x, const float* __restrict__ pesos,
    float* __restrict__ out)
{
    const int n       = blockIdx.x;        // region id, 0..1023
    const int i       = n >> 5;            // n / GW
    const int j       = n & 31;            // n % GW
    const int wave    = threadIdx.x >> 5;  // 0..7
    const int lane    = threadIdx.x & 31;
    const int lane_lo = lane & 15;
    const int hi      = lane >> 4;         // which K-half this lane holds
    const int k0      = hi * 2;            // first K element in v0

    // ---- A fragments: xd tiles (16 batches x K=4), one per b-tile ----
    // lanes 0-15: M=lane, K=0,1 ; lanes 16-31: M=lane-16, K=2,3
    const int xbase = i * (R_ * W_) + j * R_;        // i*512 + j*4
    v2f afrag[4];
#pragma unroll
    for (int tb = 0; tb < 4; ++tb) {
        const int b = tb * 16 + lane_lo;
        const float* xp = x + (size_t)b * (H_ * W_) + xbase;
        afrag[tb].x = xp[(k0 + 0) * (W_ + 1)];       // diag stride = 129
        afrag[tb].y = xp[(k0 + 1) * (W_ + 1)];
    }

    // ---- this wave's 4 ds-tiles ----
    const float* wp = pesos + (size_t)n * (DS_ * R_ * R_);   // n*8192
#pragma unroll
    for (int tt = 0; tt < 4; ++tt) {
        const int t  = wave * 4 + tt;                // ds-tile 0..31
        const int ds = t * 16 + lane_lo;             // column index

        // B fragment (4x16): VGPR v, lanes0-15 -> K=v ; lanes16-31 -> K=v+2
        v2f bfrag;
        bfrag.x = wp[ds * 16 + (k0 + 0) * 5];
        bfrag.y = wp[ds * 16 + (k0 + 1) * 5];

#pragma unroll
        for (int tb = 0; tb < 4; ++tb) {
            v8f acc = {0.f, 0.f, 0.f, 0.f, 0.f, 0.f, 0.f, 0.f};
            // 8 args: (neg_a, A, neg_b, B, c_mod, C, reuse_a, reuse_b)
            acc = __builtin_amdgcn_wmma_f32_16x16x4_f32(
                false, afrag[tb], false, bfrag, (short)0, acc, false, false);

            // D layout: VGPR v, lanes0-15: M=v ; lanes16-31: M=v+8 ; N=lane_lo
            // b = tb*16 + (v + hi*8), out idx = b*OUTROW_ + n*512 + ds
            float* op = out + (size_t)(tb * 16 + hi * 8) * OUTROW_
                            + (size_t)n * DS_ + t * 16 + lane_lo;
#pragma unroll
            for (int v = 0; v < 8; ++v) {
                // streaming output (written once, never re-read): non-temporal
                __builtin_nontemporal_store(acc[v], op + (size_t)v * OUTROW_);
            }
        }
    }
}

extern "C" void kernel_launch(void* const* d_in, const int* in_sizes, int n_in,
                              void* d_out, int out_size, void* d_ws, size_t ws_size,
                              hipStream_t stream) {
    const float* x     = (const float*)d_in[0];   // [64, 1, 128, 128] f32
    const float* pesos = (const float*)d_in[1];   // [1024, 16, 32, 4, 4] f32
    float* out = (float*)d_out;                   // [64, 1024, 16, 32] f32

    conexao_wmma_kernel<<<dim3(N_), dim3(256), 0, stream>>>(x, pesos, out);
}